// PtModule_76166950027695
// MI455X (gfx1250) — compile-verified
//
#include <hip/hip_runtime.h>

// y = 2*x + 2 over 32*2048*1024 f32 = 512 MiB total traffic.
// Pure HBM-bandwidth-bound: ~23 us floor at 23.3 TB/s on MI455X.
// Main kernel is branch-free full tiles: 8x global_load_b128 (NT, shared base
// address + immediate offsets), 16x v_pk_fma_f32, 8x global_store_b128 (NT).
// NT hints because the 512 MiB footprint exceeds the 192 MB L2 with zero reuse.
// Ragged remainder handled by a tiny scalar tail kernel (not launched for the
// reference shape, which divides evenly).

typedef float v4f __attribute__((ext_vector_type(4)));
typedef float v2f __attribute__((ext_vector_type(2)));

#define BLOCK  256
#define UNROLL 8
#define CHUNK  (BLOCK * UNROLL)   // float4s per block = 2048 (32 KiB in, 32 KiB out)

__global__ __launch_bounds__(BLOCK) void stream_fma2_full(
    const v4f* __restrict__ x, v4f* __restrict__ y) {
  const v2f k2 = {2.0f, 2.0f};  // B and C operands of v_pk_fma_f32 (hoisted)

  // Launch guarantees this block owns a complete chunk -> no bounds checks.
  unsigned base = blockIdx.x * (unsigned)CHUNK + threadIdx.x;
  const v4f* __restrict__ xp = x + base;
  v4f* __restrict__ yp = y + base;

  v4f r[UNROLL];

  // 8 loads in flight; lane-adjacent threads hit adjacent 16B (512B/wave32).
  // Offsets i*256 float4 = i*4096 B fit the 24-bit immediate offset field.
#pragma unroll
  for (int i = 0; i < UNROLL; ++i)
    r[i] = __builtin_nontemporal_load(xp + i * BLOCK);

#pragma unroll
  for (int i = 0; i < UNROLL; ++i) {
    v2f lo = {r[i].x, r[i].y};
    v2f hi = {r[i].z, r[i].w};
    asm("v_pk_fma_f32 %0, %1, %2, %3" : "=v"(lo) : "v"(lo), "v"(k2), "v"(k2));
    asm("v_pk_fma_f32 %0, %1, %2, %3" : "=v"(hi) : "v"(hi), "v"(k2), "v"(k2));
    v4f out;
    out.x = lo.x; out.y = lo.y; out.z = hi.x; out.w = hi.y;
    __builtin_nontemporal_store(out, yp + i * BLOCK);
  }
}

// Scalar tail for the ragged end (elements [start, n)). Not launched for the
// reference shape (2^26 elements divide evenly into full chunks).
__global__ __launch_bounds__(BLOCK) void stream_fma2_tail(
    const float* __restrict__ x, float* __restrict__ y,
    long long start, long long n) {
  long long i = start + (long long)blockIdx.x * BLOCK + threadIdx.x;
  if (i < n) y[i] = __builtin_fmaf(x[i], 2.0f, 2.0f);
}

extern "C" void kernel_launch(void* const* d_in, const int* in_sizes, int n_in,
                              void* d_out, int out_size, void* d_ws, size_t ws_size,
                              hipStream_t stream) {
  const float* x = (const float*)d_in[0];
  float* y = (float*)d_out;
  long long n  = (long long)in_sizes[0];   // 67,108,864 for the reference shape
  long long n4 = n >> 2;                   // full float4s

  long long full_blocks = n4 / CHUNK;      // 8192 for this shape
  long long covered = full_blocks * (long long)CHUNK * 4;  // elements handled branch-free

  if (full_blocks > 0) {
    stream_fma2_full<<<(int)full_blocks, BLOCK, 0, stream>>>(
        (const v4f*)x, (v4f*)y);
  }

  long long rem = n - covered;             // 0 for the reference shape
  if (rem > 0) {
    int tail_blocks = (int)((rem + BLOCK - 1) / BLOCK);
    stream_fma2_tail<<<tail_blocks, BLOCK, 0, stream>>>(x, y, covered, n);
  }
}